// RankingCNN_69398081569192
// MI455X (gfx1250) — compile-verified
//
#include <hip/hip_runtime.h>
#include <hip/hip_bf16.h>

#define NSUB  3
#define BATCH 32
#define CF    1024
#define KDIM  3072           // 3*32*32
#define SPA   49             // 7*7
#define HW    224

typedef __attribute__((ext_vector_type(2))) float v2f;
typedef __attribute__((ext_vector_type(8))) float v8f;
typedef __attribute__((ext_vector_type(4))) float f4;

// ---------------------------------------------------------------------------
// Kernel 1: stride-32/kernel-32 conv as a patch GEMM on V_WMMA_F32_16X16X4_F32
//   D[M=1568, N=1024] = A[M, K=3072] * Wb^T + bias,  per branch.
//   Block: 256 thr (8 wave32) = 2(M) x 4(N) waves.
//   Block tile: 32(M) x 256(N); wave tile 16 x 64 (4 v8f accumulators).
//   K-chunk 64.  B kept k-contiguous in LDS so every fragment is one b64.
// ---------------------------------------------------------------------------
__global__ __launch_bounds__(256) void conv_wmma_kernel(
    const float* __restrict__ x,      // [32,3,224,224]
    const float* __restrict__ Wb,     // [3,1024,3072]
    const float* __restrict__ bb,     // [3,1024]
    float* __restrict__ feats_out)    // [3][32,1024,7,7]
{
    const int NT = 4;                  // 1024 / 256 N-tiles
    const int branch = blockIdx.y;
    const int mt = blockIdx.x / NT;    // 0..48
    const int nt = blockIdx.x % NT;    // 0..3
    const int mBase = mt * 32;
    const int nBase = nt * 256;

    // stride 68 floats: rows 16B-aligned (f4 stores) and 4-bank rotation
    __shared__ float lA[32][68];       // A tile  [m][k]
    __shared__ float lBT[256][68];     // B tile  [n][k]  (k-contiguous!)

    const int tid  = threadIdx.x;
    const int lane = tid & 31;
    const int wave = tid >> 5;         // 8 waves
    const int wm   = wave >> 2;        // 0..1  (M sub-tile)
    const int wn   = wave & 3;         // 0..3  (N stripe of 64)
    const int half = lane >> 4;        // K-pair select per WMMA layout
    const int l16  = lane & 15;

    const float* Wbi = Wb + (size_t)branch * CF * KDIM;

    v8f acc0 = {0.f,0.f,0.f,0.f,0.f,0.f,0.f,0.f};
    v8f acc1 = acc0, acc2 = acc0, acc3 = acc0;

    for (int kBase = 0; kBase < KDIM; kBase += 64) {
        // ---- A tile: 32 rows x 64 k, gathered from x patches (2 x float4/thr)
        #pragma unroll
        for (int it = 0; it < 2; ++it) {
            int q   = tid + it * 256;          // 0..511 vec4 slots
            int row = q >> 4;                  // 0..31
            int kv  = (q & 15) << 2;           // 0..60, step 4
            int m   = mBase + row;
            int b   = m / SPA;
            int s   = m - b * SPA;
            int oh  = s / 7, ow = s - oh * 7;
            int k   = kBase + kv;
            int c   = k >> 10;                 // /1024
            int rem = k & 1023;
            int kh  = rem >> 5, kw = rem & 31; // kw % 4 == 0 -> float4 safe
            const float* src = x +
                (((size_t)(b * 3 + c) * HW + (oh * 32 + kh)) * HW + (ow * 32 + kw));
            f4 v = *(const f4*)src;
            *(f4*)&lA[row][kv] = v;
        }
        // ---- B tile: 256 n x 64 k, direct k-contiguous copy (16 x float4/thr)
        #pragma unroll
        for (int it = 0; it < 16; ++it) {
            int q  = tid + it * 256;           // 0..4095 vec4 slots
            int nl = q >> 4;                   // 0..255
            int kv = (q & 15) << 2;            // 0..60
            const float* src = Wbi + (size_t)(nBase + nl) * KDIM + kBase + kv;
            f4 v = *(const f4*)src;
            *(f4*)&lBT[nl][kv] = v;
        }
        __syncthreads();

        const int ar  = wm * 16 + l16;         // A row for this lane
        const int bc0 = wn * 64 + l16;         // B cols for the 4 sub-tiles
        #pragma unroll
        for (int kk = 0; kk < 64; kk += 4) {
            const int ks = kk + 2 * half;      // K pair start for this lane half
            // A 16x4: lanes 0-15 hold K{0,1}, lanes 16-31 K{2,3}; one b64 each.
            v2f a  = *(const v2f*)&lA[ar][ks];
            v2f b0 = *(const v2f*)&lBT[bc0     ][ks];
            v2f b1 = *(const v2f*)&lBT[bc0 + 16][ks];
            v2f b2 = *(const v2f*)&lBT[bc0 + 32][ks];
            v2f b3 = *(const v2f*)&lBT[bc0 + 48][ks];
            acc0 = __builtin_amdgcn_wmma_f32_16x16x4_f32(false, a, false, b0,
                                                         (short)0, acc0, false, false);
            acc1 = __builtin_amdgcn_wmma_f32_16x16x4_f32(false, a, false, b1,
                                                         (short)0, acc1, false, false);
            acc2 = __builtin_amdgcn_wmma_f32_16x16x4_f32(false, a, false, b2,
                                                         (short)0, acc2, false, false);
            acc3 = __builtin_amdgcn_wmma_f32_16x16x4_f32(false, a, false, b3,
                                                         (short)0, acc3, false, false);
        }
        __syncthreads();
    }

    // ---- epilogue: +bias, scatter per C/D layout into feats [b][n][s]
    float* dst = feats_out + (size_t)branch * BATCH * CF * SPA;
    v8f accs[4] = {acc0, acc1, acc2, acc3};
    #pragma unroll
    for (int nn = 0; nn < 4; ++nn) {
        const int ng = nBase + wn * 64 + nn * 16 + l16;
        const float bias = bb[branch * CF + ng];
        #pragma unroll
        for (int r = 0; r < 8; ++r) {
            int mg = mBase + wm * 16 + r + 8 * half;  // VGPR r: M=r (+8 hi lanes)
            int b  = mg / SPA;
            int s  = mg - b * SPA;
            dst[(size_t)(b * CF + ng) * SPA + s] = accs[nn][r] + bias;
        }
    }
}

// ---------------------------------------------------------------------------
// Kernel 2: GAP -> Linear(1024,512)+ReLU -> Linear(512,2) -> softmax[:,1]
//   One block per (branch, batch image).
// ---------------------------------------------------------------------------
__global__ __launch_bounds__(256) void head_kernel(
    const float* __restrict__ feats,  // [3][32,1024,49]
    const float* __restrict__ p1w,    // [3,512,1024]
    const float* __restrict__ p1b,    // [3,512]
    const float* __restrict__ p2w,    // [3,2,512]
    const float* __restrict__ p2b,    // [3,2]
    float* __restrict__ ranking)      // [32,3]
{
    const int branch = blockIdx.x / BATCH;
    const int b      = blockIdx.x % BATCH;
    const int tid    = threadIdx.x;

    __shared__ float pooled[CF];
    __shared__ float hbuf[512];
    __shared__ float red[2][256];

    const float* f = feats + (size_t)branch * BATCH * CF * SPA + (size_t)b * CF * SPA;
    for (int c = tid; c < CF; c += 256) {
        float s = 0.f;
        const float* fc = f + (size_t)c * SPA;
        #pragma unroll
        for (int p = 0; p < SPA; ++p) s += fc[p];
        pooled[c] = s * (1.0f / SPA);
    }
    __syncthreads();

    const float* w1 = p1w + (size_t)branch * 512 * CF;
    for (int j = tid; j < 512; j += 256) {
        float d = p1b[branch * 512 + j];
        const float* wr = w1 + (size_t)j * CF;
        for (int c = 0; c < CF; ++c) d += wr[c] * pooled[c];
        hbuf[j] = d > 0.f ? d : 0.f;
    }
    __syncthreads();

    const float* w2 = p2w + (size_t)branch * 2 * 512;
    float s0 = 0.f, s1 = 0.f;
    for (int j = tid; j < 512; j += 256) {
        float hv = hbuf[j];
        s0 += w2[j] * hv;
        s1 += w2[512 + j] * hv;
    }
    red[0][tid] = s0; red[1][tid] = s1;
    __syncthreads();
    for (int off = 128; off > 0; off >>= 1) {
        if (tid < off) {
            red[0][tid] += red[0][tid + off];
            red[1][tid] += red[1][tid + off];
        }
        __syncthreads();
    }
    if (tid == 0) {
        float l0 = red[0][0] + p2b[branch * 2 + 0];
        float l1 = red[1][0] + p2b[branch * 2 + 1];
        float m  = fmaxf(l0, l1);
        float e0 = __expf(l0 - m), e1 = __expf(l1 - m);
        ranking[b * NSUB + branch] = e1 / (e0 + e1);
    }
}

// ---------------------------------------------------------------------------
// Kernel 3: CAM: relu( cls_w[o,:] . feats[b,:,s] ) for o in {0,1}, s in 0..48
//   One block per (branch, batch image); one lane per (o,s) output.
// ---------------------------------------------------------------------------
__global__ __launch_bounds__(128) void cam_kernel(
    const float* __restrict__ feats,  // [3][32,1024,49]
    const float* __restrict__ clsw,   // [3,2,1024]
    float* __restrict__ cam)          // [3][2][32,49]
{
    const int branch = blockIdx.x / BATCH;
    const int b      = blockIdx.x % BATCH;
    const int tid    = threadIdx.x;
    if (tid >= 2 * SPA) return;
    const int o = tid / SPA;
    const int s = tid - o * SPA;

    const float* f = feats + (size_t)branch * BATCH * CF * SPA
                           + (size_t)b * CF * SPA + s;
    const float* w = clsw + (size_t)branch * 2 * CF + (size_t)o * CF;
    float d = 0.f;
    for (int c = 0; c < CF; ++c) d += w[c] * f[(size_t)c * SPA];
    cam[(size_t)branch * 2 * BATCH * SPA + (size_t)o * BATCH * SPA + b * SPA + s]
        = fmaxf(d, 0.f);
}

// ---------------------------------------------------------------------------
extern "C" void kernel_launch(void* const* d_in, const int* in_sizes, int n_in,
                              void* d_out, int out_size, void* d_ws, size_t ws_size,
                              hipStream_t stream) {
    const float* x    = (const float*)d_in[0];   // [32,3,224,224]
    const float* Wb   = (const float*)d_in[1];   // [3,1024,3,32,32]
    const float* bb   = (const float*)d_in[2];   // [3,1024]
    const float* p1w  = (const float*)d_in[3];   // [3,512,1024]
    const float* p1b  = (const float*)d_in[4];   // [3,512]
    const float* p2w  = (const float*)d_in[5];   // [3,2,512]
    const float* p2b  = (const float*)d_in[6];   // [3,2]
    const float* clsw = (const float*)d_in[7];   // [3,2,1024]

    float* out     = (float*)d_out;
    float* ranking = out;                                   // 32*3
    float* cam     = out + BATCH * NSUB;                    // 3*2*32*49
    float* feats   = cam + NSUB * 2 * BATCH * SPA;          // 3*32*1024*49

    dim3 g1(49 * 4, NSUB);   // 49 M-tiles x 4 N-tiles, branch on y
    conv_wmma_kernel<<<g1, 256, 0, stream>>>(x, Wb, bb, feats);
    head_kernel<<<NSUB * BATCH, 256, 0, stream>>>(feats, p1w, p1b, p2w, p2b, ranking);
    cam_kernel<<<NSUB * BATCH, 128, 0, stream>>>(feats, clsw, cam);
}